// FastFuzzySphere_16681652977957
// MI455X (gfx1250) — compile-verified
//
#include <hip/hip_runtime.h>
#include <math.h>

// FastFuzzySphere binning for MI455X (gfx1250).
//
// Shapes (from reference): B=32, N=16384, M=8192, K=32.
// Roofline: 512MB write-once output + 67MB read-once input streams ->
// ~25us floor at 23.3 TB/s HBM; ~0.5 GFLOP of transcendental VALU sits
// below that. Pure pointwise gather/bin: WMMA structurally inapplicable
// (no contraction dim). Optimize the CDNA5 memory path instead:
//   - lane==k (K==wave32): coalesced b32 streams, broadcast query loads
//   - th:NT load/store hints: read-once/write-once streams bypass L2
//     residency (output is 2.7x the 192MB L2 and never re-read)
//   - global_prefetch_b8 on the gathered database line
//   - 32-bit element offsets -> SADDR-form global addressing (no 64-bit
//     VALU address chains)
//   - all independent loads issued before the first dependent wait

namespace {
constexpr int  kN   = 16384;               // database points per batch
constexpr int  kMSh = 13;                  // M = 8192 = 2^13
constexpr int  kNAz = 4, kNEl = 2, kNRa = 2;
constexpr float kPi = 3.14159265358979323846f;
constexpr float kRadiusInv = 1.0f / 0.05f;

typedef int   vi4 __attribute__((ext_vector_type(4)));
typedef float vf4 __attribute__((ext_vector_type(4)));
}

__global__ __launch_bounds__(256, 8)
void fuzzy_sphere_bins(const float* __restrict__ database,
                       const float* __restrict__ query,
                       const int*   __restrict__ nn_index,
                       const float* __restrict__ nn_dist,
                       int*   __restrict__ out_idx,
                       float* __restrict__ out_w,
                       int total)
{
    const int e = blockIdx.x * blockDim.x + threadIdx.x;  // (b*M + m)*K + k
    if (e >= total) return;

    const int bm = e >> 5;                 // (b*M + m); lane id == k
    const int b  = bm >> kMSh;

    // ---- Issue ALL independent loads first (idx, dist, query) so the
    // clause has 4+ loads in flight before the first dependent wait. ----
    const int   idx = __builtin_nontemporal_load(nn_index + e);
    const float d   = __builtin_nontemporal_load(nn_dist  + e);
    // Query point: identical address across the 32 lanes -> one request,
    // L2-resident (3MB total).
    const float qx = query[bm * 3 + 0];
    const float qy = query[bm * 3 + 1];
    const float qz = query[bm * 3 + 2];

    // ---- Dependent gather: 32-bit element offset keeps addressing in the
    // SADDR + voffset form. Prefetch pulls the line while qx/qy/qz land. ----
    const int goff = (b * kN + idx) * 3;   // < 1.6M elements, fits 32-bit
    const float* __restrict__ np = database + goff;
    __builtin_prefetch(np, 0, 0);          // global_prefetch_b8
    const float nx = np[0];
    const float ny = np[1];
    const float nz = np[2];

    const float x = nx - qx;
    const float y = ny - qy;
    const float z = nz - qz;

    const float azimuth   = atan2f(y, x) + kPi;                           // [0, 2pi]
    const float cosel     = fminf(fmaxf(z / (d + 1e-8f), -1.0f), 1.0f);
    const float elevation = acosf(cosel);                                 // [0, pi]
    const float radial    = d * kRadiusInv;

    const float a_bin = azimuth   * ((float)kNAz / (2.0f * kPi));
    const float e_bin = elevation * ((float)kNEl / kPi);
    const float r_bin = fminf(fmaxf(radial, 0.0f), (float)kNRa - 1e-6f);

    const float a_fl = floorf(a_bin), e_fl = floorf(e_bin), r_fl = floorf(r_bin);
    const int   a_i = (int)a_fl, e_i = (int)e_fl, r_i = (int)r_fl;
    const float a_fr = a_bin - a_fl, e_fr = e_bin - e_fl, r_fr = r_bin - r_fl;

    const float wa[2] = {1.0f - a_fr, a_fr};
    const float we[2] = {1.0f - e_fr, e_fr};
    const float wr[2] = {1.0f - r_fr, r_fr};
    const int   ia[2] = { a_i & (kNAz - 1), (a_i + 1) & (kNAz - 1) };     // wrap azimuth
    const int   ie[2] = { min(max(e_i,     0), kNEl - 1),
                          min(max(e_i + 1, 0), kNEl - 1) };
    const int   ir[2] = { min(max(r_i,     0), kNRa - 1),
                          min(max(r_i + 1, 0), kNRa - 1) };

    int   iv[8];
    float wv[8];
    int c = 0;
#pragma unroll
    for (int dr = 0; dr < 2; ++dr)
#pragma unroll
        for (int de = 0; de < 2; ++de)
#pragma unroll
            for (int da = 0; da < 2; ++da) {
                iv[c] = (ia[da] * kNEl + ie[de]) * kNRa + ir[dr];
                wv[c] = wa[da] * we[de] * wr[dr];
                ++c;
            }

    const vi4 i0 = {iv[0], iv[1], iv[2], iv[3]};
    const vi4 i1 = {iv[4], iv[5], iv[6], iv[7]};
    const vf4 w0 = {wv[0], wv[1], wv[2], wv[3]};
    const vf4 w1 = {wv[4], wv[5], wv[6], wv[7]};

    // 512MB written once, never re-read: NT b128 stores keep it out of L2.
    // Each lane writes 2x16B per array; a wave covers contiguous 1KB bursts.
    const long long o = (long long)e * 8;
    __builtin_nontemporal_store(i0, (vi4*)(out_idx + o));
    __builtin_nontemporal_store(i1, (vi4*)(out_idx + o + 4));
    __builtin_nontemporal_store(w0, (vf4*)(out_w   + o));
    __builtin_nontemporal_store(w1, (vf4*)(out_w   + o + 4));
}

extern "C" void kernel_launch(void* const* d_in, const int* in_sizes, int n_in,
                              void* d_out, int out_size, void* d_ws, size_t ws_size,
                              hipStream_t stream) {
    // setup_inputs order: database, query, nn_index, nn_count, nn_dist
    const float* database = (const float*)d_in[0];
    const float* query    = (const float*)d_in[1];
    const int*   nn_index = (const int*)d_in[2];
    // d_in[3] = nn_count: unused by the reference computation (dead input)
    const float* nn_dist  = (const float*)d_in[4];

    const int total = in_sizes[2];                       // B*M*K = 8,388,608

    // Tuple output, concatenated flat in return order:
    //   [0,          total*8) : indices (int32 bit pattern)
    //   [total*8, 2*total*8)  : weights (float32)
    int*   out_idx = (int*)d_out;
    float* out_w   = (float*)d_out + (long long)total * 8;

    const int block = 256;                               // 8 waves / block
    const int grid  = (total + block - 1) / block;       // 32768 blocks
    fuzzy_sphere_bins<<<grid, block, 0, stream>>>(
        database, query, nn_index, nn_dist, out_idx, out_w, total);
}